// FixedScalarGCN_19344532702051
// MI455X (gfx1250) — compile-verified
//
#include <hip/hip_runtime.h>

// ---------------------------------------------------------------------------
// GCN forward on MI455X (gfx1250), fp32 end-to-end.
//   h0   = x @ W1 + b1                          (WMMA f32 16x16x4)
//   acc1 = spmm(adj, h0)                        (atomic scatter, L2-resident)
//   acc2 = spmm(adj, elu(acc1))
//   acc3 = spmm(adj, elu(acc2))
//   out  = elu(acc3) @ Wout + bout              (WMMA f32 16x16x4, ELU fused)
// ---------------------------------------------------------------------------

typedef __attribute__((ext_vector_type(2))) float v2f;
typedef __attribute__((ext_vector_type(8))) float v8f;

#define N_NODES 50000
#define N_EDGES 800000
#define NFEAT   256
#define NHID    96
#define NOUT    64

__device__ __forceinline__ float elu_f(float x) {
    return x > 0.0f ? x : (__expf(x) - 1.0f);
}

// ---------------------------------------------------------------------------
// Dense GEMM + bias: D[M,N] = op(A)[M,K] @ B[K,N] + bias[N]
// One wave per 16x16 output tile, V_WMMA_F32_16X16X4_F32, K stepped by 4.
//
// fp32 WMMA VGPR layouts (ISA 7.12.2):
//   A 16x4 : lanes 0-15 -> M=lane, {V0,V1} = K=k0+0,k0+1
//            lanes 16-31 -> M=lane-16, {V0,V1} = K=k0+2,k0+3
//   B 4x16 : lanes 0-15 -> N=lane, {V0,V1} = K=k0+0,k0+1
//            lanes 16-31 -> N=lane-16, {V0,V1} = K=k0+2,k0+3
//   C/D    : VGPR v: lanes 0-15 -> (M=v, N=lane); lanes 16-31 -> (M=v+8, N=lane-16)
// ---------------------------------------------------------------------------
template <int K, int LDA, int LDB, bool ELU_A>
__global__ void wmma_gemm_bias(const float* __restrict__ A,
                               const float* __restrict__ B,
                               const float* __restrict__ bias,
                               float* __restrict__ D,
                               int mTiles, int nTiles, int ldd) {
    const int wavesPerBlock = blockDim.x >> 5;
    const int wid = blockIdx.x * wavesPerBlock + (threadIdx.x >> 5);
    if (wid >= mTiles * nTiles) return;   // wave-uniform: EXEC stays all-1s

    const int mt   = wid / nTiles;
    const int nt   = wid % nTiles;
    const int lane = threadIdx.x & 31;
    const int half = lane >> 4;           // 0: K pair {0,1}, 1: K pair {2,3}
    const int lm   = lane & 15;

    const int colN = nt * 16 + lm;        // N index for B loads / D stores
    const float* aPtr = A + (size_t)(mt * 16 + lm) * LDA + 2 * half;
    const float* bPtr = B + (size_t)(2 * half) * LDB + colN;

    v8f acc = {};
    #pragma unroll 4
    for (int k0 = 0; k0 < K; k0 += 4) {
        const float* ap = aPtr + k0;
        v2f a;
        a.x = ap[0];
        a.y = ap[1];
        if (ELU_A) { a.x = elu_f(a.x); a.y = elu_f(a.y); }

        const float* bp = bPtr + (size_t)k0 * LDB;
        v2f b;
        b.x = bp[0];
        b.y = bp[LDB];

        acc = __builtin_amdgcn_wmma_f32_16x16x4_f32(
            /*neg_a=*/false, a, /*neg_b=*/false, b,
            /*c_mod=*/(short)0, acc, /*reuse_a=*/false, /*reuse_b=*/false);
    }

    const float bv = bias[colN];
    float* dBase = D + (size_t)(mt * 16 + half * 8) * ldd + colN;
    #pragma unroll
    for (int v = 0; v < 8; ++v) {
        dBase[(size_t)v * ldd] = acc[v] + bv;
    }
}

// ---------------------------------------------------------------------------
// SpMM scatter: out[row[e], :] += val[e] * (ELU?)(H[col[e], :])
// One wave per edge; NHID=96 -> 3 features per lane. H and out are L2-resident
// (19.2 MB each vs 192 MB L2), so gathers hit L2 and float atomics resolve
// at the L2 atomic units.
// ---------------------------------------------------------------------------
template <bool ELU_SRC>
__global__ void spmm_scatter(const int* __restrict__ rows,
                             const int* __restrict__ cols,
                             const float* __restrict__ vals,
                             const float* __restrict__ H,
                             float* __restrict__ out,
                             int nEdges) {
    const int e = (int)((blockIdx.x * blockDim.x + threadIdx.x) >> 5);
    if (e >= nEdges) return;
    const int lane = threadIdx.x & 31;

    const int   r = rows[e];
    const int   c = cols[e];
    const float v = vals[e];

    const float* src = H   + (size_t)c * NHID;
    float*       dst = out + (size_t)r * NHID;

    #pragma unroll
    for (int j = 0; j < 3; ++j) {
        const int f = lane + 32 * j;
        float x = src[f];
        if (ELU_SRC) x = elu_f(x);
        atomicAdd(dst + f, v * x);
    }
}

__global__ void zero_f32(float* __restrict__ p, int n) {
    const int i = blockIdx.x * blockDim.x + threadIdx.x;
    if (i < n) p[i] = 0.0f;
}

extern "C" void kernel_launch(void* const* d_in, const int* in_sizes, int n_in,
                              void* d_out, int out_size, void* d_ws, size_t ws_size,
                              hipStream_t stream) {
    (void)in_sizes; (void)n_in; (void)out_size; (void)ws_size;

    const float* x       = (const float*)d_in[0];
    const int*   adj_row = (const int*)  d_in[1];
    const int*   adj_col = (const int*)  d_in[2];
    const float* adj_val = (const float*)d_in[3];
    const float* W1      = (const float*)d_in[4];
    const float* b1      = (const float*)d_in[5];
    const float* Wout    = (const float*)d_in[6];
    const float* bout    = (const float*)d_in[7];
    float*       out     = (float*)d_out;

    float* bufA = (float*)d_ws;                       // [N_NODES, NHID]
    float* bufB = bufA + (size_t)N_NODES * NHID;      // [N_NODES, NHID]

    const int HN = N_NODES * NHID;                    // 4.8M elements
    const int zeroBlocks = (HN + 255) / 256;
    const int spmmBlocks = (N_EDGES + 7) / 8;         // 8 waves (edges) / 256-thr block

    // --- GEMM1: bufA = x @ W1 + b1 ---------------------------------------
    {
        const int mTiles = N_NODES / 16;              // 3125
        const int nTiles = NHID / 16;                 // 6
        const int waves  = mTiles * nTiles;
        const int wpb    = 4;                         // 128-thread blocks
        const int blocks = (waves + wpb - 1) / wpb;
        wmma_gemm_bias<NFEAT, NFEAT, NHID, false>
            <<<blocks, wpb * 32, 0, stream>>>(x, W1, b1, bufA, mTiles, nTiles, NHID);
    }

    // --- Layer 0: acc1 = spmm(h0) ----------------------------------------
    zero_f32<<<zeroBlocks, 256, 0, stream>>>(bufB, HN);
    spmm_scatter<false><<<spmmBlocks, 256, 0, stream>>>(adj_row, adj_col, adj_val,
                                                        bufA, bufB, N_EDGES);
    // --- Layer 1: acc2 = spmm(elu(acc1)) ---------------------------------
    zero_f32<<<zeroBlocks, 256, 0, stream>>>(bufA, HN);
    spmm_scatter<true><<<spmmBlocks, 256, 0, stream>>>(adj_row, adj_col, adj_val,
                                                       bufB, bufA, N_EDGES);
    // --- Layer 2: acc3 = spmm(elu(acc2)) ---------------------------------
    zero_f32<<<zeroBlocks, 256, 0, stream>>>(bufB, HN);
    spmm_scatter<true><<<spmmBlocks, 256, 0, stream>>>(adj_row, adj_col, adj_val,
                                                       bufA, bufB, N_EDGES);

    // --- GEMM2: out = elu(acc3) @ Wout + bout (ELU fused into A-load) ----
    {
        const int mTiles = N_NODES / 16;              // 3125
        const int nTiles = NOUT / 16;                 // 4
        const int waves  = mTiles * nTiles;
        const int wpb    = 4;
        const int blocks = (waves + wpb - 1) / wpb;
        wmma_gemm_bias<NHID, NHID, NOUT, true>
            <<<blocks, wpb * 32, 0, stream>>>(bufB, Wout, bout, out, mTiles, nTiles, NOUT);
    }
}